// RobustGCNConv_9440338117105
// MI455X (gfx1250) — compile-verified
//
#include <hip/hip_runtime.h>
#include <hip/hip_bf16.h>

typedef __attribute__((ext_vector_type(2))) float v2f;
typedef __attribute__((ext_vector_type(8))) float v8f;

#define IN_F   256
#define OUT_F  256
#define LDS_STRIDE 260   // pad 256 -> 260 floats: lane L bank = (4L + k) % 64, conflict-free

// ---------------------------------------------------------------------------
// Kernel 1: fused dense layer for both mean and var paths + activation fusion
//   m = elu(mean @ Wm + bm);  v = relu(var @ Wv + bv)
//   att = exp(-v);  m_ws = m*att;  v_ws = v*att*att
// One block = 16 nodes. 8 waves; each wave computes two 16x16 col-tiles for
// BOTH matrices via V_WMMA_F32_16X16X4_F32 (exact f32 path, K stepped by 4).
// ---------------------------------------------------------------------------
__global__ __launch_bounds__(256) void dense_attn_kernel(
    const float* __restrict__ mean, const float* __restrict__ var,
    const float* __restrict__ Wm,   const float* __restrict__ bm,
    const float* __restrict__ Wv,   const float* __restrict__ bv,
    float* __restrict__ m_ws, float* __restrict__ v_ws)
{
  __shared__ __align__(16) float ldsA[2][16][LDS_STRIDE];

  const int rowbase = blockIdx.x * 16;
  const int tid = threadIdx.x;

  // Cooperative, coalesced load of the two 16x256 A tiles into LDS.
  for (int i = tid; i < 16 * (IN_F / 4); i += 256) {
    const int rr  = i / (IN_F / 4);
    const int cc4 = i % (IN_F / 4);
    const float4 am = ((const float4*)(mean + (size_t)(rowbase + rr) * IN_F))[cc4];
    const float4 av = ((const float4*)(var  + (size_t)(rowbase + rr) * IN_F))[cc4];
    *(float4*)&ldsA[0][rr][cc4 * 4] = am;   // row stride 1040 B = 65*16, 16B aligned
    *(float4*)&ldsA[1][rr][cc4 * 4] = av;
  }
  __syncthreads();

  const int wid   = tid >> 5;        // wave 0..7
  const int lane  = tid & 31;
  const int lhalf = lane & 15;
  const int khalf = (lane >> 4) * 2; // lanes 0-15 carry K+0/K+1, lanes 16-31 carry K+2/K+3

  const int colb[2] = { wid * 16, (wid + 8) * 16 };

  v8f accM0 = {}, accM1 = {}, accV0 = {}, accV1 = {};

  for (int k = 0; k < IN_F; k += 4) {
    v2f aM, aV;
    aM.x = ldsA[0][lhalf][k + khalf];
    aM.y = ldsA[0][lhalf][k + khalf + 1];
    aV.x = ldsA[1][lhalf][k + khalf];
    aV.y = ldsA[1][lhalf][k + khalf + 1];

    const size_t r0 = (size_t)(k + khalf) * OUT_F;
    const size_t r1 = (size_t)(k + khalf + 1) * OUT_F;

    {
      const int col = colb[0] + lhalf;
      v2f bM, bV;
      bM.x = Wm[r0 + col]; bM.y = Wm[r1 + col];
      bV.x = Wv[r0 + col]; bV.y = Wv[r1 + col];
      accM0 = __builtin_amdgcn_wmma_f32_16x16x4_f32(false, aM, false, bM,
                                                    (short)0, accM0, false, false);
      accV0 = __builtin_amdgcn_wmma_f32_16x16x4_f32(false, aV, false, bV,
                                                    (short)0, accV0, false, false);
    }
    {
      const int col = colb[1] + lhalf;
      v2f bM, bV;
      bM.x = Wm[r0 + col]; bM.y = Wm[r1 + col];
      bV.x = Wv[r0 + col]; bV.y = Wv[r1 + col];
      accM1 = __builtin_amdgcn_wmma_f32_16x16x4_f32(false, aM, false, bM,
                                                    (short)0, accM1, false, false);
      accV1 = __builtin_amdgcn_wmma_f32_16x16x4_f32(false, aV, false, bV,
                                                    (short)0, accV1, false, false);
    }
  }

  // Epilogue: bias + elu/relu + attention, fully in registers.
  const int mbase = (lane >> 4) * 8;
  for (int t = 0; t < 2; ++t) {
    const int col = colb[t] + lhalf;
    const float biasM = bm[col];
    const float biasV = bv[col];
    const v8f accM = t ? accM1 : accM0;
    const v8f accV = t ? accV1 : accV0;
#pragma unroll
    for (int i = 0; i < 8; ++i) {
      float mval = accM[i] + biasM;
      float vval = accV[i] + biasV;
      mval = (mval > 0.0f) ? mval : (__expf(mval) - 1.0f);   // elu
      vval = (vval > 0.0f) ? vval : 0.0f;                    // relu
      const float att = __expf(-vval);
      const int node  = rowbase + mbase + i;
      m_ws[(size_t)node * OUT_F + col] = mval * att;
      v_ws[(size_t)node * OUT_F + col] = vval * att * att;
    }
  }
}

// ---------------------------------------------------------------------------
// CSR build: histogram rows, exclusive scan, bucket fill.
// ---------------------------------------------------------------------------
__global__ __launch_bounds__(256) void count_kernel(
    const int* __restrict__ erow, int* __restrict__ cnt, int n_edges)
{
  const int e = blockIdx.x * blockDim.x + threadIdx.x;
  if (e < n_edges) atomicAdd(&cnt[erow[e]], 1);
}

__global__ __launch_bounds__(1024) void scan_kernel(
    const int* __restrict__ cnt, int* __restrict__ offs,
    int* __restrict__ cursor, int n)
{
  __shared__ int buf[1024];
  __shared__ int carry_s;
  const int tid = threadIdx.x;
  if (tid == 0) carry_s = 0;
  __syncthreads();

  for (int base = 0; base < n; base += 1024) {
    const int i = base + tid;
    const int x = (i < n) ? cnt[i] : 0;
    buf[tid] = x;
    __syncthreads();
#pragma unroll
    for (int off = 1; off < 1024; off <<= 1) {
      const int y = (tid >= off) ? buf[tid - off] : 0;
      __syncthreads();
      buf[tid] += y;
      __syncthreads();
    }
    const int incl  = buf[tid];
    const int carry = carry_s;
    if (i < n) {
      offs[i]   = carry + incl - x;   // exclusive
      cursor[i] = carry + incl - x;
    }
    __syncthreads();
    if (tid == 1023) carry_s = carry + incl;
    __syncthreads();
  }
  if (tid == 0) offs[n] = carry_s;
}

__global__ __launch_bounds__(256) void fill_kernel(
    const int* __restrict__ erow, int* __restrict__ cursor,
    int* __restrict__ eidx, int n_edges)
{
  const int e = blockIdx.x * blockDim.x + threadIdx.x;
  if (e < n_edges) {
    const int pos = atomicAdd(&cursor[erow[e]], 1);
    eidx[pos] = e;
  }
}

// ---------------------------------------------------------------------------
// Kernel 2: gather-form dual SpMM. One wave per output node; accumulate its
// 256-float m and v rows in registers over its CSR edge list; single b128
// stores at the end. No float atomics anywhere.
// ---------------------------------------------------------------------------
__global__ __launch_bounds__(256) void spmm_gather_kernel(
    const int* __restrict__ offs, const int* __restrict__ eidx,
    const int* __restrict__ ecol,
    const float* __restrict__ a0, const float* __restrict__ a1,
    const float* __restrict__ m_ws, const float* __restrict__ v_ws,
    float* __restrict__ out_m, float* __restrict__ out_v, int n_nodes)
{
  const int wid  = threadIdx.x >> 5;
  const int lane = threadIdx.x & 31;
  const int node = blockIdx.x * 8 + wid;
  if (node >= n_nodes) return;

  const int beg = offs[node];
  const int end = offs[node + 1];

  float m0x = 0.f, m0y = 0.f, m0z = 0.f, m0w = 0.f;
  float m1x = 0.f, m1y = 0.f, m1z = 0.f, m1w = 0.f;
  float v0x = 0.f, v0y = 0.f, v0z = 0.f, v0w = 0.f;
  float v1x = 0.f, v1y = 0.f, v1z = 0.f, v1w = 0.f;

  for (int j = beg; j < end; ++j) {
    const int   e  = eidx[j];
    const int   c  = ecol[e];
    const float w0 = a0[e];
    const float w1 = a1[e];
    const float4* __restrict__ ms = (const float4*)(m_ws + (size_t)c * OUT_F);
    const float4* __restrict__ vs = (const float4*)(v_ws + (size_t)c * OUT_F);
    const float4 mA = ms[lane];
    const float4 mB = ms[lane + 32];
    const float4 vA = vs[lane];
    const float4 vB = vs[lane + 32];
    m0x = fmaf(w0, mA.x, m0x); m0y = fmaf(w0, mA.y, m0y);
    m0z = fmaf(w0, mA.z, m0z); m0w = fmaf(w0, mA.w, m0w);
    m1x = fmaf(w0, mB.x, m1x); m1y = fmaf(w0, mB.y, m1y);
    m1z = fmaf(w0, mB.z, m1z); m1w = fmaf(w0, mB.w, m1w);
    v0x = fmaf(w1, vA.x, v0x); v0y = fmaf(w1, vA.y, v0y);
    v0z = fmaf(w1, vA.z, v0z); v0w = fmaf(w1, vA.w, v0w);
    v1x = fmaf(w1, vB.x, v1x); v1y = fmaf(w1, vB.y, v1y);
    v1z = fmaf(w1, vB.z, v1z); v1w = fmaf(w1, vB.w, v1w);
  }

  float4* __restrict__ md = (float4*)(out_m + (size_t)node * OUT_F);
  float4* __restrict__ vd = (float4*)(out_v + (size_t)node * OUT_F);
  md[lane]      = make_float4(m0x, m0y, m0z, m0w);
  md[lane + 32] = make_float4(m1x, m1y, m1z, m1w);
  vd[lane]      = make_float4(v0x, v0y, v0z, v0w);
  vd[lane + 32] = make_float4(v1x, v1y, v1z, v1w);
}

extern "C" void kernel_launch(void* const* d_in, const int* in_sizes, int n_in,
                              void* d_out, int out_size, void* d_ws, size_t ws_size,
                              hipStream_t stream) {
  const float* mean = (const float*)d_in[0];
  const float* var  = (const float*)d_in[1];
  const float* Wm   = (const float*)d_in[2];
  const float* bm   = (const float*)d_in[3];
  const float* Wv   = (const float*)d_in[4];
  const float* bv   = (const float*)d_in[5];
  const int*   erow = (const int*)d_in[6];
  const int*   ecol = (const int*)d_in[7];
  const float* a0   = (const float*)d_in[8];
  const float* a1   = (const float*)d_in[9];

  const int n_edges = in_sizes[6];
  const int n_nodes = in_sizes[0] / IN_F;   // 100000

  // Workspace layout:
  //   m_ws : n_nodes*256 f32      (102.4 MB)
  //   v_ws : n_nodes*256 f32      (102.4 MB)
  //   cnt  : n_nodes i32
  //   offs : n_nodes+1 i32
  //   cur  : n_nodes i32
  //   eidx : n_edges i32
  float* m_ws = (float*)d_ws;
  float* v_ws = m_ws + (size_t)n_nodes * OUT_F;
  int*   cnt  = (int*)(v_ws + (size_t)n_nodes * OUT_F);
  int*   offs = cnt  + n_nodes;
  int*   cur  = offs + (n_nodes + 1);
  int*   eidx = cur  + n_nodes;

  float* out_m = (float*)d_out;
  float* out_v = out_m + (size_t)n_nodes * OUT_F;

  // 1) dense + activation + attention (WMMA f32)
  dense_attn_kernel<<<n_nodes / 16, 256, 0, stream>>>(
      mean, var, Wm, bm, Wv, bv, m_ws, v_ws);

  // 2) CSR build
  hipMemsetAsync(cnt, 0, (size_t)n_nodes * sizeof(int), stream);
  const int eblocks = (n_edges + 255) / 256;
  count_kernel<<<eblocks, 256, 0, stream>>>(erow, cnt, n_edges);
  scan_kernel<<<1, 1024, 0, stream>>>(cnt, offs, cur, n_nodes);
  fill_kernel<<<eblocks, 256, 0, stream>>>(erow, cur, eidx, n_edges);

  // 3) gather-form dual SpMM (no float atomics, single store per row)
  const int nblocks = (n_nodes + 7) / 8;
  spmm_gather_kernel<<<nblocks, 256, 0, stream>>>(
      offs, eidx, ecol, a0, a1, m_ws, v_ws, out_m, out_v, n_nodes);
}